// GNN_65369402245874
// MI455X (gfx1250) — compile-verified
//
#include <hip/hip_runtime.h>
#include <hip/hip_bf16.h>

// ---- problem constants (match reference) ----
#define BN   8
#define NN   128
#define DN   768
#define EN   64
#define MLN  128
#define SN   64
#define HN   128
#define NHN  12

typedef float v2f __attribute__((ext_vector_type(2)));
typedef float v8f __attribute__((ext_vector_type(8)));
typedef unsigned int u32x4 __attribute__((ext_vector_type(4)));
typedef int i32x4 __attribute__((ext_vector_type(4)));
typedef int i32x8 __attribute__((ext_vector_type(8)));

#if defined(__has_builtin)
#if __has_builtin(__builtin_amdgcn_tensor_load_to_lds) && \
    __has_builtin(__builtin_amdgcn_s_wait_tensorcnt)
#define HAVE_TDM 1
#endif
#endif
#ifndef HAVE_TDM
#define HAVE_TDM 0
#endif

// f32 WMMA: D(16x16) = A(16x4) * B(4x16) + C   (CDNA5 V_WMMA_F32_16X16X4_F32)
// args: (neg_a, A, neg_b, B, c_mod, C, reuse_a, reuse_b)
__device__ __forceinline__ v8f wmma4(v2f a, v2f b, v8f c) {
  return __builtin_amdgcn_wmma_f32_16x16x4_f32(false, a, false, b, (short)0, c,
                                               false, false);
}

// LDS strides (floats), padded for bank-conflict-free row access:
#define LXS 68    // 64-wide edge tile rows (64 data + 4 pad DWORDs)
#define LTS 130   // 128-wide activation rows

#if HAVE_TDM
// Tensor Data Mover: DMA a (128 x 64) fp32 tile, contiguous in global
// (row stride 64 elements), into LDS with +4-DWORD padding per 64-DWORD row
// (reproduces the LXS=68 layout). D# per CDNA5 ISA.
// Issue from ONE wave only (EXEC is ignored; every issuing wave would DMA).
__device__ __forceinline__ void tdm_load_edge_tile(const float* gsrc,
                                                   unsigned lds_byte_addr) {
  const unsigned long long ga = (unsigned long long)(uintptr_t)gsrc;
  u32x4 g0;
  g0[0] = 1u;                                   // count=1 (valid), no gather
  g0[1] = lds_byte_addr;                        // lds_addr [63:32]
  g0[2] = (unsigned)ga;                         // global_addr [95:64]
  g0[3] = (unsigned)((ga >> 32) & 0x01FFFFFFu)  // global_addr [120:96]
          | (2u << 30);                         // type=2 ("image")
  i32x8 g1;
  g1[0] = (int)((2u << 16)     // data_size = 4B
              | (1u << 20)     // pad_enable
              | (5u << 22)     // pad_interval: 2^(5+1)=64 DWORDs between pads
              | (3u << 25));   // pad_amount: 3+1 = 4 DWORDs of padding
  g1[1] = (int)(64u << 16);    // tensor_dim0 = 64 elements  (bits 79:48)
  g1[2] = (int)(128u << 16);   // tensor_dim1 = 128 rows     (bits 111:80)
  g1[3] = (int)(64u << 16);    // tile_dim0 = 64             (bits 127:112)
  g1[4] = (int)128u;           // tile_dim1 = 128 rows       (bits 143:128)
  g1[5] = (int)64u;            // tensor_dim0_stride = 64    (bits 207:160)
  g1[6] = (int)(8192u << 16);  // tensor_dim1_stride = 8192  (bits 255:208)
  g1[7] = 0;
  i32x4 gz4 = {0, 0, 0, 0};    // groups 2/3 unused (2D tile)
#if defined(__clang_major__) && (__clang_major__ >= 23)
  i32x8 gz8 = {0, 0, 0, 0, 0, 0, 0, 0};
  __builtin_amdgcn_tensor_load_to_lds(g0, g1, gz4, gz4, gz8, 0);
#else
  __builtin_amdgcn_tensor_load_to_lds(g0, g1, gz4, gz4, 0);
#endif
}
#endif

// ---------------------------------------------------------------------------
// Generic row-major GEMM: C[M,N] = act(A[M,K] @ W[K,N]), ldW == N.
// 128 rows per block, 8 waves; wave owns 16-col strips; 8 m-tiles per strip.
// ---------------------------------------------------------------------------
__global__ __launch_bounds__(256) void k_gemm(const float* __restrict__ A,
                                              const float* __restrict__ W,
                                              float* __restrict__ C,
                                              int K, int N, int do_relu) {
  const int wave = threadIdx.x >> 5;
  const int lane = threadIdx.x & 31;
  const int half = lane >> 4;
  const int lr   = lane & 15;
  const int m0   = blockIdx.x * 128;

  for (int nt = wave; nt * 16 < N; nt += 8) {
    const int n0 = nt * 16;
    v8f acc[8] = {};
    for (int k0 = 0; k0 < K; k0 += 4) {
      const int kk = k0 + 2 * half;
      v2f bf;
      bf.x = W[(size_t)kk * N + n0 + lr];
      bf.y = W[(size_t)(kk + 1) * N + n0 + lr];
#pragma unroll
      for (int mt = 0; mt < 8; ++mt) {
        v2f af = *reinterpret_cast<const v2f*>(
            &A[(size_t)(m0 + mt * 16 + lr) * K + kk]);
        acc[mt] = wmma4(af, bf, acc[mt]);
      }
    }
#pragma unroll
    for (int mt = 0; mt < 8; ++mt)
#pragma unroll
      for (int r = 0; r < 8; ++r) {
        const int row = m0 + mt * 16 + r + 8 * half;
        float v = acc[mt][r];
        if (do_relu) v = fmaxf(v, 0.0f);
        C[(size_t)row * N + n0 + lr] = v;
      }
  }
}

// ---------------------------------------------------------------------------
// LDS(128x128,pad LTS) @ W(128xldw) -> relu -> LDS(128x128,pad LTS)
// ---------------------------------------------------------------------------
__device__ __forceinline__ void gemm128_lds_relu(const float* __restrict__ Asrc,
                                                 const float* __restrict__ W,
                                                 int ldw,
                                                 float* __restrict__ Dst,
                                                 int n0, int half, int lr) {
  v8f acc[8] = {};
  for (int k0 = 0; k0 < 128; k0 += 4) {
    const int kk = k0 + 2 * half;
    v2f bf;
    bf.x = W[kk * ldw + n0 + lr];
    bf.y = W[(kk + 1) * ldw + n0 + lr];
#pragma unroll
    for (int mt = 0; mt < 8; ++mt) {
      v2f af = *reinterpret_cast<const v2f*>(&Asrc[(mt * 16 + lr) * LTS + kk]);
      acc[mt] = wmma4(af, bf, acc[mt]);
    }
  }
#pragma unroll
  for (int mt = 0; mt < 8; ++mt)
#pragma unroll
    for (int r = 0; r < 8; ++r) {
      const int row = mt * 16 + r + 8 * half;
      Dst[row * LTS + n0 + lr] = fmaxf(acc[mt][r], 0.0f);
    }
}

// X(128x64 LDS) @ We(64x128) + self[h] + neigh[j,h] -> relu -> Dst (LDS)
__device__ __forceinline__ void stage_edge_in(const float* __restrict__ lx,
                                              const float* __restrict__ We,
                                              const float* __restrict__ vself,
                                              const float* __restrict__ vneigh,
                                              int b, int bi,
                                              float* __restrict__ Dst,
                                              int n0, int half, int lr) {
  v8f acc[8] = {};
  for (int k0 = 0; k0 < 64; k0 += 4) {
    const int kk = k0 + 2 * half;
    v2f bf;
    bf.x = We[kk * HN + n0 + lr];
    bf.y = We[(kk + 1) * HN + n0 + lr];
#pragma unroll
    for (int mt = 0; mt < 8; ++mt) {
      v2f af = *reinterpret_cast<const v2f*>(&lx[(mt * 16 + lr) * LXS + kk]);
      acc[mt] = wmma4(af, bf, acc[mt]);
    }
  }
  const float sv = vself[(size_t)bi * HN + n0 + lr];
#pragma unroll
  for (int mt = 0; mt < 8; ++mt)
#pragma unroll
    for (int r = 0; r < 8; ++r) {
      const int row = mt * 16 + r + 8 * half;  // neighbor index j
      float v = acc[mt][r] + sv +
                vneigh[((size_t)(b * NN + row)) * HN + n0 + lr];
      Dst[row * LTS + n0 + lr] = fmaxf(v, 0.0f);
    }
}

// ---------------------------------------------------------------------------
// Hot kernel: one workgroup per (b,i). Runs message-gen MLP (+ mean over j)
// and edge-update MLP entirely out of LDS. Edge tile staged by the TDM.
// ---------------------------------------------------------------------------
__global__ __launch_bounds__(256) void k_edge_msg(
    const float* __restrict__ edge_in,                    // (B,N,N,E)
    const float* __restrict__ a_self,  const float* __restrict__ a_neigh,
    const float* __restrict__ e_self,  const float* __restrict__ e_neigh,
    const float* __restrict__ m_w1e,   const float* __restrict__ m_w2,
    const float* __restrict__ m_w3,
    const float* __restrict__ e_w1e,   const float* __restrict__ e_w2,
    const float* __restrict__ e_w3,
    float* __restrict__ message,                          // (B,N,ML)
    float* __restrict__ edge_out) {                       // (B,N,N,E)
  __shared__ float lx[128 * LXS];   // edge tile (128 x 64, +4 pad/row)
  __shared__ float lt[128 * LTS];   // activation ping
  __shared__ float lu[128 * LTS];   // activation pong

  const int bi   = blockIdx.x;      // b*N + i
  const int b    = bi >> 7;
  const int wave = threadIdx.x >> 5;
  const int lane = threadIdx.x & 31;
  const int half = lane >> 4;
  const int lr   = lane & 15;
  const int n0   = wave * 16;

  // ---- stage edge tile (32KB) into LDS
#if HAVE_TDM
  if (wave == 0) {
    tdm_load_edge_tile(edge_in + (size_t)bi * (NN * EN),
                       (unsigned)(uintptr_t)&lx[0]);
    __builtin_amdgcn_s_wait_tensorcnt(0);
  }
#else
  {
    const float4* src =
        reinterpret_cast<const float4*>(edge_in + (size_t)bi * (NN * EN));
#pragma unroll
    for (int it = 0; it < 8; ++it) {
      const int idx4 = threadIdx.x + it * 256;  // 2048 float4s
      const float4 v = src[idx4];
      const int row = idx4 >> 4;
      const int col = (idx4 & 15) * 4;
      *reinterpret_cast<float4*>(&lx[row * LXS + col]) = v;
    }
  }
#endif
  __syncthreads();

  // ---- S1: lt = relu(X @ m_w1[2D:] + a_self_i + a_neigh_j)
  stage_edge_in(lx, m_w1e, a_self, a_neigh, b, bi, lt, n0, half, lr);
  __syncthreads();

  // ---- S2: lu = relu(lt @ m_w2)
  gemm128_lds_relu(lt, m_w2, HN, lu, n0, half, lr);
  __syncthreads();

  // ---- S3: message[b,i,:] = mean_j (lu @ m_w3)  (column sums in-register)
  {
    v8f acc[8] = {};
    for (int k0 = 0; k0 < 128; k0 += 4) {
      const int kk = k0 + 2 * half;
      v2f bf;
      bf.x = m_w3[kk * MLN + n0 + lr];
      bf.y = m_w3[(kk + 1) * MLN + n0 + lr];
#pragma unroll
      for (int mt = 0; mt < 8; ++mt) {
        v2f af = *reinterpret_cast<const v2f*>(&lu[(mt * 16 + lr) * LTS + kk]);
        acc[mt] = wmma4(af, bf, acc[mt]);
      }
    }
    float cs = 0.0f;
#pragma unroll
    for (int mt = 0; mt < 8; ++mt)
#pragma unroll
      for (int r = 0; r < 8; ++r) cs += acc[mt][r];
    cs += __shfl_xor(cs, 16, 32);  // combine the two row-halves (wave32)
    if (half == 0)
      message[(size_t)bi * MLN + n0 + lr] = cs * (1.0f / 128.0f);
  }

  // ---- S4: lt = relu(X @ e_w1[2S:] + e_self_i + e_neigh_j)
  stage_edge_in(lx, e_w1e, e_self, e_neigh, b, bi, lt, n0, half, lr);
  __syncthreads();  // covers S3's lu reads and S4's lt writes

  // ---- S5: lu = relu(lt @ e_w2)
  gemm128_lds_relu(lt, e_w2, HN, lu, n0, half, lr);
  __syncthreads();

  // ---- S6: new_edge = lu @ e_w3 (128->64); edge_out = X + new_edge*(X>0)
  {
    const int ct = wave & 3;        // column tile (4 tiles of 16 over E=64)
    const int rh = wave >> 2;       // row half (rows 0..63 / 64..127)
    const int e0 = ct * 16;
    v8f acc[4] = {};
    for (int k0 = 0; k0 < 128; k0 += 4) {
      const int kk = k0 + 2 * half;
      v2f bf;
      bf.x = e_w3[kk * EN + e0 + lr];
      bf.y = e_w3[(kk + 1) * EN + e0 + lr];
#pragma unroll
      for (int mt = 0; mt < 4; ++mt) {
        const int rbase = rh * 64 + mt * 16;
        v2f af = *reinterpret_cast<const v2f*>(&lu[(rbase + lr) * LTS + kk]);
        acc[mt] = wmma4(af, bf, acc[mt]);
      }
    }
#pragma unroll
    for (int mt = 0; mt < 4; ++mt)
#pragma unroll
      for (int r = 0; r < 8; ++r) {
        const int j = rh * 64 + mt * 16 + r + 8 * half;
        const float x = lx[j * LXS + e0 + lr];
        const float v = x + (x > 0.0f ? acc[mt][r] : 0.0f);
        edge_out[((size_t)bi * NN + j) * EN + e0 + lr] = v;
      }
  }
}

// ---------------------------------------------------------------------------
// Node update: one workgroup per batch b.
// node_out = node_in + MLP(cat(message, node_in))
// ---------------------------------------------------------------------------
__global__ __launch_bounds__(256) void k_node_update(
    const float* __restrict__ node_in,   // (B,N,D)
    const float* __restrict__ message,   // (B,N,ML)
    const float* __restrict__ n_w1,      // (ML+D, H)
    const float* __restrict__ n_w2,      // (H,H)
    const float* __restrict__ n_w3,      // (H,D)
    float* __restrict__ node_out) {      // (B,N,D)
  __shared__ float lt[128 * LTS];
  __shared__ float lu[128 * LTS];

  const int b    = blockIdx.x;
  const int wave = threadIdx.x >> 5;
  const int lane = threadIdx.x & 31;
  const int half = lane >> 4;
  const int lr   = lane & 15;
  const int n0   = wave * 16;

  // ---- S1: lt = relu(msg @ n_w1[:ML] + node @ n_w1[ML:])
  {
    v8f acc[8] = {};
    for (int k0 = 0; k0 < MLN; k0 += 4) {
      const int kk = k0 + 2 * half;
      v2f bf;
      bf.x = n_w1[kk * HN + n0 + lr];
      bf.y = n_w1[(kk + 1) * HN + n0 + lr];
#pragma unroll
      for (int mt = 0; mt < 8; ++mt) {
        v2f af = *reinterpret_cast<const v2f*>(
            &message[((size_t)(b * NN + mt * 16 + lr)) * MLN + kk]);
        acc[mt] = wmma4(af, bf, acc[mt]);
      }
    }
    for (int k0 = 0; k0 < DN; k0 += 4) {
      const int kk = k0 + 2 * half;
      v2f bf;
      bf.x = n_w1[(MLN + kk) * HN + n0 + lr];
      bf.y = n_w1[(MLN + kk + 1) * HN + n0 + lr];
#pragma unroll
      for (int mt = 0; mt < 8; ++mt) {
        v2f af = *reinterpret_cast<const v2f*>(
            &node_in[((size_t)(b * NN + mt * 16 + lr)) * DN + kk]);
        acc[mt] = wmma4(af, bf, acc[mt]);
      }
    }
#pragma unroll
    for (int mt = 0; mt < 8; ++mt)
#pragma unroll
      for (int r = 0; r < 8; ++r) {
        const int row = mt * 16 + r + 8 * half;
        lt[row * LTS + n0 + lr] = fmaxf(acc[mt][r], 0.0f);
      }
  }
  __syncthreads();

  // ---- S2: lu = relu(lt @ n_w2)
  gemm128_lds_relu(lt, n_w2, HN, lu, n0, half, lr);
  __syncthreads();

  // ---- S3: node_out = node_in + lu @ n_w3   (128 -> 768 cols)
  for (int nt = wave; nt < 48; nt += 8) {
    const int c0 = nt * 16;
    v8f acc[8] = {};
    for (int k0 = 0; k0 < 128; k0 += 4) {
      const int kk = k0 + 2 * half;
      v2f bf;
      bf.x = n_w3[kk * DN + c0 + lr];
      bf.y = n_w3[(kk + 1) * DN + c0 + lr];
#pragma unroll
      for (int mt = 0; mt < 8; ++mt) {
        v2f af = *reinterpret_cast<const v2f*>(&lu[(mt * 16 + lr) * LTS + kk]);
        acc[mt] = wmma4(af, bf, acc[mt]);
      }
    }
#pragma unroll
    for (int mt = 0; mt < 8; ++mt)
#pragma unroll
      for (int r = 0; r < 8; ++r) {
        const int row = mt * 16 + r + 8 * half;
        const size_t off = ((size_t)(b * NN + row)) * DN + c0 + lr;
        node_out[off] = node_in[off] + acc[mt][r];
      }
  }
}

// ---------------------------------------------------------------------------
// Final edge_mask_layer: out(B,N,N,NH) = edge @ mask_w + mask_b (NH=12, tiny)
// ---------------------------------------------------------------------------
__global__ __launch_bounds__(256) void k_mask(const float* __restrict__ edge,
                                              const float* __restrict__ mw,
                                              const float* __restrict__ mb,
                                              float* __restrict__ out) {
  const int gid = blockIdx.x * 256 + threadIdx.x;
  if (gid >= BN * NN * NN * NHN) return;
  const int h = gid % NHN;
  const int row = gid / NHN;
  const float* er = edge + (size_t)row * EN;
  float s = mb[h];
#pragma unroll
  for (int e = 0; e < EN; ++e) s += er[e] * mw[e * NHN + h];
  out[gid] = s;
}

// ---------------------------------------------------------------------------
extern "C" void kernel_launch(void* const* d_in, const int* in_sizes, int n_in,
                              void* d_out, int out_size, void* d_ws,
                              size_t ws_size, hipStream_t stream) {
  (void)in_sizes; (void)n_in; (void)out_size; (void)ws_size;
  const float* node0  = (const float*)d_in[0];
  const float* edge0  = (const float*)d_in[1];
  // d_in[2] = num_layers (== 2 in setup; unrolled here)
  const float* s_w1   = (const float*)d_in[3];
  const float* s_w2   = (const float*)d_in[4];
  const float* m_w1   = (const float*)d_in[5];
  const float* m_w2   = (const float*)d_in[6];
  const float* m_w3   = (const float*)d_in[7];
  const float* n_w1   = (const float*)d_in[8];
  const float* n_w2   = (const float*)d_in[9];
  const float* n_w3   = (const float*)d_in[10];
  const float* e_w1   = (const float*)d_in[11];
  const float* e_w2   = (const float*)d_in[12];
  const float* e_w3   = (const float*)d_in[13];
  const float* mask_w = (const float*)d_in[14];
  const float* mask_b = (const float*)d_in[15];

  float* ws      = (float*)d_ws;
  float* a_self  = ws;                       // 1024*128
  float* a_neigh = a_self  + 131072;
  float* shrink  = a_neigh + 131072;         // 1024*64
  float* tmpS    = shrink  + 65536;          // 1024*128
  float* e_self  = tmpS    + 131072;
  float* e_neigh = e_self  + 131072;
  float* msg     = e_neigh + 131072;         // 1024*128
  float* node1   = msg     + 131072;         // 8*128*768
  float* edge1   = node1   + 786432;         // 8*128*128*64
  float* edge2   = edge1   + 8388608;

  const float* node_ins[2]  = {node0, node1};
  const float* edge_ins[2]  = {edge0, edge1};
  float*       node_outs[2] = {node1, (float*)d_out};
  float*       edge_outs[2] = {edge1, edge2};

  const dim3 blk(256);
  for (int L = 0; L < 2; ++L) {
    const float* nin = node_ins[L];
    const float* ein = edge_ins[L];
    // per-node precompute GEMMs (M = 1024 rows -> 8 blocks each)
    k_gemm<<<dim3(8), blk, 0, stream>>>(nin, m_w1,            a_self,  DN, HN, 0);
    k_gemm<<<dim3(8), blk, 0, stream>>>(nin, m_w1 + DN * HN,  a_neigh, DN, HN, 0);
    k_gemm<<<dim3(8), blk, 0, stream>>>(nin, s_w1,            tmpS,    DN, HN, 1);
    k_gemm<<<dim3(8), blk, 0, stream>>>(tmpS, s_w2,           shrink,  HN, SN, 0);
    k_gemm<<<dim3(8), blk, 0, stream>>>(shrink, e_w1,            e_self,  SN, HN, 0);
    k_gemm<<<dim3(8), blk, 0, stream>>>(shrink, e_w1 + SN * HN,  e_neigh, SN, HN, 0);
    // hot per-edge kernel: 1024 workgroups (one per (b,i))
    k_edge_msg<<<dim3(BN * NN), blk, 0, stream>>>(
        ein, a_self, a_neigh, e_self, e_neigh,
        m_w1 + 2 * DN * HN, m_w2, m_w3,
        e_w1 + 2 * SN * HN, e_w2, e_w3,
        msg, edge_outs[L]);
    // node update
    k_node_update<<<dim3(BN), blk, 0, stream>>>(nin, msg, n_w1, n_w2, n_w3,
                                                node_outs[L]);
  }
  // final mask layer on post-layer-2 edges -> second half of d_out
  k_mask<<<dim3((BN * NN * NN * NHN + 255) / 256), blk, 0, stream>>>(
      edge2, mask_w, mask_b, (float*)d_out + (size_t)BN * NN * DN);
}